// MetricsRoi3D_11141145166159
// MI455X (gfx1250) — compile-verified
//
#include <hip/hip_runtime.h>
#include <math.h>

// Problem constants (from reference setup_inputs)
#define BD 128
#define HD 256
#define WD 256
#define NC 4
#define KK 10
#define PB 4            // (K-1)/2 pad-begin; pad-end = 5
#define TILE_H 16
#define TROWS (TILE_H + KK - 1)   // 25 rows incl. halo
#define TSTRIDE 272               // 265 cols padded to 16B-friendly stride
#define NBINS 10
#define NCHAN 112                 // 101 channels used, padded to 7*16 for WMMA batches
#define NWAVES 8

typedef __attribute__((ext_vector_type(2))) float v2f;
typedef __attribute__((ext_vector_type(8))) float v8f;

#if __has_builtin(__builtin_amdgcn_global_load_async_to_lds_b128)
#define HAVE_ASYNC_LDS 1
// Builtin signature (from hipcc diagnostic): (int __vector(4) AS1*, int __vector(4) AS3*, Ii, Ii)
typedef int vi4 __attribute__((__vector_size__(16)));
typedef __attribute__((address_space(1))) vi4 gvi4;   // global
typedef __attribute__((address_space(3))) vi4 lvi4;   // LDS
#endif

__device__ __forceinline__ void stage_b128(const int* __restrict__ src, int* lds_dst) {
#if defined(HAVE_ASYNC_LDS)
  // async DMA: global -> LDS, tracked by ASYNCcnt (no VGPR round trip)
  __builtin_amdgcn_global_load_async_to_lds_b128((gvi4*)src, (lvi4*)lds_dst,
                                                 /*imm offset*/0, /*cpol*/0);
#else
  *(int4*)lds_dst = *(const int4*)src;   // fallback: global_load_b128 + ds_store_b128
#endif
}

__device__ __forceinline__ void wait_async_lds() {
#if defined(HAVE_ASYNC_LDS)
#if __has_builtin(__builtin_amdgcn_s_wait_asynccnt)
  __builtin_amdgcn_s_wait_asynccnt(0);
#else
  asm volatile("s_wait_asynccnt 0" ::: "memory");
#endif
#endif
}

// jnp.digitize(x, linspace(0,1,11), right=True) - 1, clipped to [0,9]
__device__ __forceinline__ int bin_of(float x) {
  int bi = (int)ceilf(x * 10.0f) - 1;
  bi = bi < 0 ? 0 : bi;
  return bi > (NBINS - 1) ? (NBINS - 1) : bi;
}

extern "C" __global__ void init_part(float* __restrict__ g) {
  if (threadIdx.x < NCHAN) g[threadIdx.x] = 0.0f;
}

extern "C" __global__ void __launch_bounds__(256)
metrics_main(const float* __restrict__ pred, const float* __restrict__ dw,
             const int* __restrict__ gth, float* __restrict__ gpart) {
  __shared__ int   tile[TROWS * TSTRIDE];      // label strip + halo (zero-padded)
  __shared__ float bins[NWAVES * NCHAN];       // per-wave bin accumulators

  const int tid = threadIdx.x;
  const int bb  = blockIdx.x >> 4;             // batch index (H/TILE_H == 16 strips)
  const int r0  = (blockIdx.x & 15) * TILE_H;  // strip start row
  const long long Npix = (long long)BD * HD * WD;

  // zero per-wave bins and the LDS tile (halo cells stay 0 == zero padding)
  for (int i = tid; i < NWAVES * NCHAN; i += 256) bins[i] = 0.0f;
  for (int i = tid; i < TROWS * TSTRIDE; i += 256) tile[i] = 0;
  __syncthreads();

  // ---- stage labels (with halo) global -> LDS via async b128 copies ----
  {
    const int c4  = (tid & 63) * 4;            // 64 threads x 4 ints cover one 256-wide row
    const int tr0 = tid >> 6;                  // 4 rows per iteration
#pragma unroll
    for (int it = 0; it < 7; ++it) {
      const int t  = it * 4 + tr0;             // tile row
      const int gr = r0 - PB + t;              // global row
      if (t < TROWS && gr >= 0 && gr < HD) {
        const int* src = gth + ((long long)bb * HD + gr) * WD + c4;
        int* dst = &tile[t * TSTRIDE + PB + c4];
        stage_b128(src, dst);
      }
    }
  }
  wait_async_lds();
  __syncthreads();

  // dilation weights are wave-uniform: load once, reuse for all 16 rows
  float wv[KK * KK];
#pragma unroll
  for (int i = 0; i < KK * KK; ++i) wv[i] = dw[i];

  const int col = tid;                          // blockDim.x == W == 256
  for (int rr = 0; rr < TILE_H; ++rr) {
    // grey dilation: roi = max_{di,dj} fg_pad[r+di][c+dj] + w[di][dj]
    float m = -INFINITY;
#pragma unroll
    for (int di = 0; di < KK; ++di) {
      const int* trow = &tile[(rr + di) * TSTRIDE + col];
#pragma unroll
      for (int dj = 0; dj < KK; ++dj) {
        const float fg = (trow[dj] > 0) ? 1.0f : 0.0f;  // clip(label,0,1)
        m = fmaxf(m, fg + wv[di * KK + dj]);
      }
    }

    if (m > 0.5f) {                             // eff_mask weight == 1, else contributes 0
      const long long n = ((long long)bb * HD + (r0 + rr)) * WD + col;
      const int lab = tile[(rr + PB) * TSTRIDE + PB + col];
      const float p0 = pred[n];
      const float p1 = pred[Npix + n];
      const float p2 = pred[2 * Npix + n];
      const float p3 = pred[3 * Npix + n];
      float conf = p0; int km = 0;              // first-max argmax (jnp.argmax)
      if (p1 > conf) { conf = p1; km = 1; }
      if (p2 > conf) { conf = p2; km = 2; }
      if (p3 > conf) { conf = p3; km = 3; }

      float* wb = &bins[(tid >> 5) * NCHAN];    // this wave's private bins
      atomicAdd(&wb[100], 1.0f);                // total mask weight
      const int bc = bin_of(conf);              // ECE: bin by confidence
      atomicAdd(&wb[bc],      (km == lab) ? 1.0f : 0.0f);
      atomicAdd(&wb[10 + bc], conf);
      // SCE v2: per class k, bin by probs[k]
      const int b0 = bin_of(p0);
      atomicAdd(&wb[20 + b0], (lab == 0) ? 1.0f : 0.0f); atomicAdd(&wb[30 + b0], p0);
      const int b1 = bin_of(p1);
      atomicAdd(&wb[40 + b1], (lab == 1) ? 1.0f : 0.0f); atomicAdd(&wb[50 + b1], p1);
      const int b2 = bin_of(p2);
      atomicAdd(&wb[60 + b2], (lab == 2) ? 1.0f : 0.0f); atomicAdd(&wb[70 + b2], p2);
      const int b3 = bin_of(p3);
      atomicAdd(&wb[80 + b3], (lab == 3) ? 1.0f : 0.0f); atomicAdd(&wb[90 + b3], p3);
    }
  }
  __syncthreads();

  // ---- cross-wave reduction of the 8 per-wave bin copies via f32 WMMA ----
  // D(16x16) = A(16x4) x B(4x16) + C with B == 1.0: D[m,*] = sum_k A[m,k] (exact f32).
  // A[m,k] = bins[wave k][batch*16 + m]; two chained WMMAs cover waves 0..3 and 4..7.
  // All 8 waves execute the WMMA with full EXEC (required); only wave 0 commits.
  {
    const int lane = tid & 31;
    const int m16  = lane & 15;
    const int kb   = (lane >> 4) * 2;           // A layout: lanes0-15 hold K0/K1, lanes16-31 K2/K3
#if __has_builtin(__builtin_amdgcn_wmma_f32_16x16x4_f32)
    v2f bones; bones.x = 1.0f; bones.y = 1.0f;
#pragma unroll
    for (int t = 0; t < 7; ++t) {
      v2f a0, a1;
      a0.x = bins[(kb + 0) * NCHAN + t * 16 + m16];
      a0.y = bins[(kb + 1) * NCHAN + t * 16 + m16];
      a1.x = bins[(kb + 4) * NCHAN + t * 16 + m16];
      a1.y = bins[(kb + 5) * NCHAN + t * 16 + m16];
      v8f c = {};
      c = __builtin_amdgcn_wmma_f32_16x16x4_f32(false, a0, false, bones, (short)0, c, false, false);
      c = __builtin_amdgcn_wmma_f32_16x16x4_f32(false, a1, false, bones, (short)0, c, false, false);
      // D layout: VGPR i holds M=i (lanes 0-15, N=lane) and M=i+8 (lanes 16-31, N=lane-16).
      // Column N=0 lives in lane 0 (M=0..7) and lane 16 (M=8..15). Wave 0 commits.
      if (tid == 0 || tid == 16) {
        const int mb = t * 16 + ((tid == 16) ? 8 : 0);
#pragma unroll
        for (int i = 0; i < 8; ++i) atomicAdd(&gpart[mb + i], c[i]);
      }
    }
#else
    if (tid < 32) {
      for (int ch = lane; ch < NCHAN; ch += 32) {
        float s = 0.0f;
        for (int w8 = 0; w8 < NWAVES; ++w8) s += bins[w8 * NCHAN + ch];
        atomicAdd(&gpart[ch], s);
      }
    }
#endif
  }
}

extern "C" __global__ void metrics_final(const float* __restrict__ g, float* __restrict__ out) {
  if (threadIdx.x == 0) {
    const float total = g[100];
    float ece = 0.0f;
    for (int bi = 0; bi < NBINS; ++bi) ece += fabsf(g[bi] - g[10 + bi]);
    float sce = 0.0f;
    for (int k = 0; k < NC; ++k)
      for (int bi = 0; bi < NBINS; ++bi)
        sce += fabsf(g[20 + k * 20 + bi] - g[20 + k * 20 + 10 + bi]);
    out[0] = ece / total;
    out[1] = sce / (total * (float)NC);
  }
}

extern "C" void kernel_launch(void* const* d_in, const int* in_sizes, int n_in,
                              void* d_out, int out_size, void* d_ws, size_t ws_size,
                              hipStream_t stream) {
  (void)in_sizes; (void)n_in; (void)out_size; (void)ws_size;
  const float* pred = (const float*)d_in[0];   // (4,128,256,256) f32
  const float* dw   = (const float*)d_in[1];   // (1,1,10,10)     f32
  const int*   gth  = (const int*)d_in[2];     // (128,256,256)   i32
  float* gpart = (float*)d_ws;                 // 112 f32 partials
  float* out   = (float*)d_out;                // [ece, sce]

  init_part<<<dim3(1), dim3(128), 0, stream>>>(gpart);
  metrics_main<<<dim3(BD * (HD / TILE_H)), dim3(256), 0, stream>>>(pred, dw, gth, gpart);
  metrics_final<<<dim3(1), dim3(32), 0, stream>>>(gpart, out);
}